// TrafficGNN_38087769981349
// MI455X (gfx1250) — compile-verified
//
#include <hip/hip_runtime.h>
#include <hip/hip_bf16.h>
#include <math.h>

typedef __attribute__((ext_vector_type(2))) float v2f;
typedef __attribute__((ext_vector_type(8))) float v8f;

#define FIN 128
#define HID 64

static __device__ __forceinline__ void edge_ep(int e, const int* __restrict__ ei, int E,
                                               int& s, int& d) {
  if (e < E) { s = ei[e]; d = ei[E + e]; }
  else       { s = e - E; d = e - E; }          // self-loops appended after real edges
}

// ---------------------------------------------------------------------------
// generic fill
// ---------------------------------------------------------------------------
__global__ void k_fill(float* __restrict__ p, float v, size_t n) {
  size_t i = (size_t)blockIdx.x * blockDim.x + threadIdx.x;
  size_t stride = (size_t)gridDim.x * blockDim.x;
  for (; i < n; i += stride) p[i] = v;
}

// ---------------------------------------------------------------------------
// fp32 GEMM via V_WMMA_F32_16X16X4_F32, templated on K and M so the K-loop
// fully unrolls with immediate address offsets.
//
// Block = 32 * (M/16) threads; wave w owns column tile w. Each wave preloads
// its K x 16 B-panel into 64 VGPRs (v2f breg[K/4]) once, then sweeps TILES
// row tiles of A, so the steady-state loop is: global_load_b64 (A) + v_wmma.
//
// f32 16x4 A layout: lane L(0..15) row=L holds K=kb,kb+1 ; lanes 16..31 kb=2.
// C/D: VGPR i = row i (+8 for lanes 16..31), col = lane&15.
// ---------------------------------------------------------------------------
template<int K, int M, int TILES>
__global__ __launch_bounds__(32 * (M / 16))
void k_gemm_wmma(const float* __restrict__ A, const float* __restrict__ B,
                 float* __restrict__ C, int N) {
  const int wave = threadIdx.x >> 5;
  const int lane = threadIdx.x & 31;
  const int col0 = wave * 16;
  const int r  = lane & 15;
  const int kb = (lane >> 4) << 1;              // 0 or 2
  const int mb = (lane >> 4) << 3;              // 0 or 8

  // ---- preload this wave's B panel into registers (reused for TILES tiles)
  v2f breg[K / 4];
#pragma unroll
  for (int k4 = 0; k4 < K / 4; ++k4) {
    const float* __restrict__ Bp = B + (size_t)(4 * k4 + kb) * M + col0 + r;
    breg[k4].x = Bp[0];
    breg[k4].y = Bp[M];
  }

  for (int t = 0; t < TILES; ++t) {
    const int row0 = (blockIdx.x * TILES + t) * 16;
    if (row0 >= N) return;
    int arow = row0 + r;
    if (arow >= N) arow = N - 1;                // clamp reads; writes guarded below
    const float* __restrict__ Ar = A + (size_t)arow * K + kb;

    v8f acc0 = {};
    v8f acc1 = {};
#pragma unroll
    for (int k4 = 0; k4 < K / 4; k4 += 2) {
      v2f a0 = *(const v2f*)(Ar + 4 * k4);
      v2f a1 = *(const v2f*)(Ar + 4 * k4 + 4);
      acc0 = __builtin_amdgcn_wmma_f32_16x16x4_f32(false, a0, false, breg[k4],
                                                   (short)0, acc0, false, false);
      acc1 = __builtin_amdgcn_wmma_f32_16x16x4_f32(false, a1, false, breg[k4 + 1],
                                                   (short)0, acc1, false, false);
    }
    v8f acc = acc0 + acc1;

    float* __restrict__ Cp = C + (size_t)(row0 + mb) * M + col0 + r;
#pragma unroll
    for (int i = 0; i < 8; ++i) {
      if (row0 + mb + i < N) Cp[(size_t)i * M] = acc[i];
    }
  }
}

// ---------------------------------------------------------------------------
// per-(node,head) attention coefficients: a_src = <h, att_src>, a_dst = <h, att_dst>
// one wave per (node, head); C == 64
// ---------------------------------------------------------------------------
__global__ void k_attn_coef(const float* __restrict__ h, const float* __restrict__ att_src,
                            const float* __restrict__ att_dst, float* __restrict__ as,
                            float* __restrict__ ad, int N, int H) {
  const int gw   = (int)((blockIdx.x * (size_t)blockDim.x + threadIdx.x) >> 5);
  const int lane = threadIdx.x & 31;
  if (gw >= N * H) return;
  const int node = gw / H;
  const int head = gw - node * H;
  const float* __restrict__ hp = h + ((size_t)node * H + head) * HID;
  const float* __restrict__ ws = att_src + head * HID;
  const float* __restrict__ wd = att_dst + head * HID;
  float s0 = hp[lane] * ws[lane] + hp[lane + 32] * ws[lane + 32];
  float s1 = hp[lane] * wd[lane] + hp[lane + 32] * wd[lane + 32];
#pragma unroll
  for (int off = 16; off; off >>= 1) {
    s0 += __shfl_down(s0, off, 32);
    s1 += __shfl_down(s1, off, 32);
  }
  if (lane == 0) {
    as[node * H + head] = s0;
    ad[node * H + head] = s1;
  }
}

// ---------------------------------------------------------------------------
// per-(edge,head): alpha = leaky_relu(a_src[src]+a_dst[dst]); segment max into m[dst]
// (__hip_atomic_fetch_max on float -> global_atomic_max_num_f32)
// ---------------------------------------------------------------------------
__global__ void k_edge_alpha_max(const int* __restrict__ ei, const float* __restrict__ as,
                                 const float* __restrict__ ad, float* __restrict__ m,
                                 float* __restrict__ abuf, int E, int Etot, int H) {
  int idx = (int)(blockIdx.x * (size_t)blockDim.x + threadIdx.x);
  if (idx >= Etot * H) return;
  int e  = idx / H;
  int hh = idx - e * H;
  int s, d;
  edge_ep(e, ei, E, s, d);
  float a = as[s * H + hh] + ad[d * H + hh];
  a = (a > 0.f) ? a : a * 0.2f;
  abuf[idx] = a;
  __hip_atomic_fetch_max(&m[d * H + hh], a, __ATOMIC_RELAXED, __HIP_MEMORY_SCOPE_AGENT);
}

// ---------------------------------------------------------------------------
// per-(edge,head): e = exp(alpha - m[dst]); segment sum into s[dst]; abuf <- e
// ---------------------------------------------------------------------------
__global__ void k_edge_expsum(const int* __restrict__ ei, const float* __restrict__ m,
                              float* __restrict__ abuf, float* __restrict__ ssum,
                              int E, int Etot, int H) {
  int idx = (int)(blockIdx.x * (size_t)blockDim.x + threadIdx.x);
  if (idx >= Etot * H) return;
  int e  = idx / H;
  int hh = idx - e * H;
  int s, d;
  edge_ep(e, ei, E, s, d);
  float ex = expf(abuf[idx] - m[d * H + hh]);
  abuf[idx] = ex;
  atomicAdd(&ssum[d * H + hh], ex);
}

// ---------------------------------------------------------------------------
// per-edge weighted message aggregation: out[dst] += h[src] * a   (a per head)
// one wave per edge, lanes striped over H*C channels (C == 64); fully
// coalesced 128B gathers and same-cacheline global_atomic_add_f32 scatters
// (h and out are L2-resident: 25.6 MB << 192 MB L2).
// ---------------------------------------------------------------------------
__global__ void k_edge_aggregate(const int* __restrict__ ei, const float* __restrict__ h,
                                 const float* __restrict__ abuf, const float* __restrict__ ssum,
                                 float* __restrict__ out, int E, int Etot, int H, int HC) {
  const int e    = (int)((blockIdx.x * (size_t)blockDim.x + threadIdx.x) >> 5);
  const int lane = threadIdx.x & 31;
  if (e >= Etot) return;
  int s, d;
  edge_ep(e, ei, E, s, d);
  const float* __restrict__ hs = h + (size_t)s * HC;
  float* __restrict__ od = out + (size_t)d * HC;
  __builtin_prefetch(hs + lane, 0, 1);          // global_prefetch_b8 of the gathered row
  float coef = 0.f;
  if (lane < H) coef = abuf[e * H + lane] / (ssum[d * H + lane] + 1e-16f);
  for (int f = lane; f < HC; f += 32) {
    float c = __shfl(coef, f >> 6, 32);         // head = f / 64
    atomicAdd(&od[f], hs[f] * c);
  }
}

// ---------------------------------------------------------------------------
// y += bias (in place), accumulate per-feature sum and sumsq (F is power of 2)
// ---------------------------------------------------------------------------
__global__ void k_bias_stats(float* __restrict__ y, const float* __restrict__ bias,
                             float* __restrict__ sum, float* __restrict__ sq,
                             size_t n, int F) {
  __shared__ float ls[128];
  __shared__ float lq[128];
  for (int f = threadIdx.x; f < F; f += blockDim.x) { ls[f] = 0.f; lq[f] = 0.f; }
  __syncthreads();
  const size_t Fm = (size_t)(F - 1);
  const size_t stride = (size_t)gridDim.x * blockDim.x;
  for (size_t i = (size_t)blockIdx.x * blockDim.x + threadIdx.x; i < n; i += stride) {
    int f = (int)(i & Fm);
    float v = y[i] + bias[f];
    y[i] = v;
    atomicAdd(&ls[f], v);
    atomicAdd(&lq[f], v * v);
  }
  __syncthreads();
  for (int f = threadIdx.x; f < F; f += blockDim.x) {
    atomicAdd(&sum[f], ls[f]);
    atomicAdd(&sq[f],  lq[f]);
  }
}

// ---------------------------------------------------------------------------
// per-feature BN scale/shift from accumulated stats (one block of F threads)
// ---------------------------------------------------------------------------
__global__ void k_bn_finalize(const float* __restrict__ sum, const float* __restrict__ sq,
                              const float* __restrict__ g, const float* __restrict__ be,
                              float* __restrict__ scale, float* __restrict__ shift,
                              float invN, int F) {
  int f = threadIdx.x;
  if (f < F) {
    float mu  = sum[f] * invN;
    float var = sq[f] * invN - mu * mu;          // biased var (torch training-mode)
    float sc  = g[f] * rsqrtf(var + 1e-5f);
    scale[f] = sc;
    shift[f] = be[f] - mu * sc;
  }
}

// ---------------------------------------------------------------------------
// y = relu(y*scale + shift), in place (F power of 2)
// ---------------------------------------------------------------------------
__global__ void k_bn_relu(float* __restrict__ y, const float* __restrict__ scale,
                          const float* __restrict__ shift, size_t n, int F) {
  const size_t Fm = (size_t)(F - 1);
  size_t i = (size_t)blockIdx.x * blockDim.x + threadIdx.x;
  size_t stride = (size_t)gridDim.x * blockDim.x;
  for (; i < n; i += stride) {
    int f = (int)(i & Fm);
    float v = y[i] * scale[f] + shift[f];
    y[i] = v > 0.f ? v : 0.f;
  }
}

// ---------------------------------------------------------------------------
// out[n] = <h[n,:], linW> + linb   (one wave per node, HID == 64)
// ---------------------------------------------------------------------------
__global__ void k_linear(const float* __restrict__ h, const float* __restrict__ w,
                         const float* __restrict__ b, float* __restrict__ out, int N) {
  const int gw   = (int)((blockIdx.x * (size_t)blockDim.x + threadIdx.x) >> 5);
  const int lane = threadIdx.x & 31;
  if (gw >= N) return;
  const float* __restrict__ hp = h + (size_t)gw * HID;
  float s = hp[lane] * w[lane] + hp[lane + 32] * w[lane + 32];
#pragma unroll
  for (int off = 16; off; off >>= 1) s += __shfl_down(s, off, 32);
  if (lane == 0) out[gw] = s + b[0];
}

// ---------------------------------------------------------------------------
// host launch
// ---------------------------------------------------------------------------
extern "C" void kernel_launch(void* const* d_in, const int* in_sizes, int n_in,
                              void* d_out, int out_size, void* d_ws, size_t ws_size,
                              hipStream_t stream) {
  const float* x        = (const float*)d_in[0];
  const int*   ei       = (const int*)  d_in[1];
  const float* W1       = (const float*)d_in[2];
  const float* att_src1 = (const float*)d_in[3];
  const float* att_dst1 = (const float*)d_in[4];
  const float* b1       = (const float*)d_in[5];
  const float* g1       = (const float*)d_in[6];
  const float* be1      = (const float*)d_in[7];
  const float* W2       = (const float*)d_in[8];
  const float* att_src2 = (const float*)d_in[9];
  const float* att_dst2 = (const float*)d_in[10];
  const float* b2       = (const float*)d_in[11];
  const float* g2       = (const float*)d_in[12];
  const float* be2      = (const float*)d_in[13];
  const float* linW     = (const float*)d_in[14];
  const float* linb     = (const float*)d_in[15];

  const int N    = in_sizes[0] / FIN;
  const int E    = in_sizes[1] / 2;
  const int Etot = E + N;                // + self loops
  const int F1   = 2 * HID;              // 128
  const int F2   = HID;                  // 64

  // ---- workspace carve-out (256B aligned) ----
  char* base = (char*)d_ws;
  size_t off = 0;
  auto alloc = [&](size_t elems) -> float* {
    float* p = (float*)(base + off);
    off += (elems * sizeof(float) + 255) & ~(size_t)255;
    return p;
  };
  float* h1   = alloc((size_t)N * F1);
  float* o1   = alloc((size_t)N * F1);
  float* as1  = alloc((size_t)N * 2);
  float* ad1  = alloc((size_t)N * 2);
  float* m1   = alloc((size_t)N * 2);
  float* s1   = alloc((size_t)N * 2);
  float* ab1  = alloc((size_t)Etot * 2);
  float* h2   = alloc((size_t)N * F2);
  float* o2   = alloc((size_t)N * F2);
  float* as2  = alloc((size_t)N);
  float* ad2  = alloc((size_t)N);
  float* m2   = alloc((size_t)N);
  float* s2   = alloc((size_t)N);
  float* ab2  = alloc((size_t)Etot);
  float* sum1 = alloc(128); float* sq1 = alloc(128);
  float* sc1  = alloc(128); float* sh1 = alloc(128);
  float* sum2 = alloc(64);  float* sq2 = alloc(64);
  float* sc2  = alloc(64);  float* sh2 = alloc(64);
  (void)ws_size; (void)n_in;

  const int BT = 256;
  const float NEG_INF = -3.0e38f;
  const int rowTiles  = (N + 15) / 16;
  const int gemmBlks  = (rowTiles + 3) / 4;     // 4 row tiles per block

  // ================= layer 1 =================
  k_fill<<<512, BT, 0, stream>>>(m1, NEG_INF, (size_t)N * 2);
  k_fill<<<512, BT, 0, stream>>>(s1, 0.f, (size_t)N * 2);
  k_fill<<<2048, BT, 0, stream>>>(o1, 0.f, (size_t)N * F1);
  k_fill<<<1, BT, 0, stream>>>(sum1, 0.f, 128);
  k_fill<<<1, BT, 0, stream>>>(sq1, 0.f, 128);

  k_gemm_wmma<FIN, 2 * HID, 4><<<gemmBlks, 32 * (F1 / 16), 0, stream>>>(x, W1, h1, N);

  {
    int waves = N * 2;
    k_attn_coef<<<(waves * 32 + BT - 1) / BT, BT, 0, stream>>>(h1, att_src1, att_dst1,
                                                               as1, ad1, N, 2);
  }
  {
    int nb = (Etot * 2 + BT - 1) / BT;
    k_edge_alpha_max<<<nb, BT, 0, stream>>>(ei, as1, ad1, m1, ab1, E, Etot, 2);
    k_edge_expsum  <<<nb, BT, 0, stream>>>(ei, m1, ab1, s1, E, Etot, 2);
  }
  k_edge_aggregate<<<((size_t)Etot * 32 + BT - 1) / BT, BT, 0, stream>>>(
      ei, h1, ab1, s1, o1, E, Etot, 2, F1);

  k_bias_stats<<<1024, BT, 0, stream>>>(o1, b1, sum1, sq1, (size_t)N * F1, F1);
  k_bn_finalize<<<1, 128, 0, stream>>>(sum1, sq1, g1, be1, sc1, sh1, 1.f / (float)N, F1);
  k_bn_relu<<<2048, BT, 0, stream>>>(o1, sc1, sh1, (size_t)N * F1, F1);

  // ================= layer 2 =================
  k_fill<<<512, BT, 0, stream>>>(m2, NEG_INF, (size_t)N);
  k_fill<<<512, BT, 0, stream>>>(s2, 0.f, (size_t)N);
  k_fill<<<2048, BT, 0, stream>>>(o2, 0.f, (size_t)N * F2);
  k_fill<<<1, BT, 0, stream>>>(sum2, 0.f, 64);
  k_fill<<<1, BT, 0, stream>>>(sq2, 0.f, 64);

  k_gemm_wmma<2 * HID, HID, 4><<<gemmBlks, 32 * (F2 / 16), 0, stream>>>(o1, W2, h2, N);

  {
    int waves = N;
    k_attn_coef<<<(waves * 32 + BT - 1) / BT, BT, 0, stream>>>(h2, att_src2, att_dst2,
                                                               as2, ad2, N, 1);
  }
  {
    int nb = (Etot + BT - 1) / BT;
    k_edge_alpha_max<<<nb, BT, 0, stream>>>(ei, as2, ad2, m2, ab2, E, Etot, 1);
    k_edge_expsum  <<<nb, BT, 0, stream>>>(ei, m2, ab2, s2, E, Etot, 1);
  }
  k_edge_aggregate<<<((size_t)Etot * 32 + BT - 1) / BT, BT, 0, stream>>>(
      ei, h2, ab2, s2, o2, E, Etot, 1, F2);

  k_bias_stats<<<1024, BT, 0, stream>>>(o2, b2, sum2, sq2, (size_t)N * F2, F2);
  k_bn_finalize<<<1, 64, 0, stream>>>(sum2, sq2, g2, be2, sc2, sh2, 1.f / (float)N, F2);
  k_bn_relu<<<2048, BT, 0, stream>>>(o2, sc2, sh2, (size_t)N * F2, F2);

  // ================= head =================
  k_linear<<<((size_t)N * 32 + BT - 1) / BT, BT, 0, stream>>>(o2, linW, linb,
                                                              (float*)d_out, N);
}